// RayleighKLLoss_41790031790568
// MI455X (gfx1250) — compile-verified
//
#include <hip/hip_runtime.h>
#include <hip/hip_bf16.h>

#define NBINS 50
#define KEPS 1e-8f
#define CH_STRIDE (512u * 512u)   // H*W = 262144 = 2^18
#define LOG2_CH 18
#define SAMPLE_STRIDE (2u * 512u * 512u)

typedef __attribute__((ext_vector_type(2))) float v2f;
typedef __attribute__((ext_vector_type(8))) float v8f;

// ---------------------------------------------------------------------------
// Sum 64 values held 2-per-lane across one wave32, via V_WMMA_F32_16X16X4_F32.
// B (4x16 f32) layout per ISA: lane n<16: {B[0][n],B[1][n]}, lane n>=16:
// {B[2][n-16],B[3][n-16]}. With A = ones(16x4), D[m][n] = sum_k B[k][n], so
// c[0] in every lane holds a column sum; 4 xor-shuffles inside each 16-lane
// half produce the grand total in all lanes.
// ---------------------------------------------------------------------------
__device__ __forceinline__ float wave_sum64_wmma(float vx, float vy) {
    v2f a = {1.0f, 1.0f};
    v2f b = {vx, vy};
    v8f c = {};
    c = __builtin_amdgcn_wmma_f32_16x16x4_f32(
        /*neg_a=*/false, a, /*neg_b=*/false, b,
        /*c_mod=*/(short)0, c, /*reuse_a=*/false, /*reuse_b=*/false);
    float s = c[0];
    s += __shfl_xor(s, 8, 32);
    s += __shfl_xor(s, 4, 32);
    s += __shfl_xor(s, 2, 32);
    s += __shfl_xor(s, 1, 32);
    return s;   // total in every lane
}

// ws layout (uint32): [0]=min bits, [1]=max bits, [2..51]=bin counts
__global__ void k_init(unsigned int* __restrict__ ws) {
    int t = threadIdx.x;
    if (t == 0) ws[0] = 0x7F7FFFFFu;   // +FLT_MAX bits
    if (t == 1) ws[1] = 0u;            // max init (all r > 0)
    if (t < NBINS) ws[2 + t] = 0u;
}

__device__ __forceinline__ float4 load_r4(const float* __restrict__ y,
                                          unsigned int i, unsigned int stride) {
    unsigned int g = i << 2;                       // pixel id (x4 vectorized)
    unsigned int bsamp = g >> LOG2_CH;             // sample index
    unsigned int p = g & (CH_STRIDE - 1u);         // pixel within sample
    const float* base = y + (size_t)bsamp * SAMPLE_STRIDE + p;
    float4 x0 = *(const float4*)(base);
    float4 x1 = *(const float4*)(base + CH_STRIDE);
    // gfx1250 prefetch of next grid-stride iteration (speculative; OOB safe)
    __builtin_prefetch((const char*)base + (size_t)stride * 16u, 0, 1);
    __builtin_prefetch((const char*)(base + CH_STRIDE) + (size_t)stride * 16u, 0, 1);
    float4 r;
    r.x = fmaxf(sqrtf(x0.x * x0.x + x1.x * x1.x), 1e-6f);
    r.y = fmaxf(sqrtf(x0.y * x0.y + x1.y * x1.y), 1e-6f);
    r.z = fmaxf(sqrtf(x0.z * x0.z + x1.z * x1.z), 1e-6f);
    r.w = fmaxf(sqrtf(x0.w * x0.w + x1.w * x1.w), 1e-6f);
    return r;
}

__global__ __launch_bounds__(256) void k_minmax(const float* __restrict__ y,
                                                unsigned int* __restrict__ ws,
                                                unsigned int total4) {
    unsigned int tid = blockIdx.x * blockDim.x + threadIdx.x;
    unsigned int stride = gridDim.x * blockDim.x;
    float mn = 3.402823466e38f;
    float mx = 0.0f;
    for (unsigned int i = tid; i < total4; i += stride) {
        float4 r = load_r4(y, i, stride);
        mn = fminf(mn, fminf(fminf(r.x, r.y), fminf(r.z, r.w)));
        mx = fmaxf(mx, fmaxf(fmaxf(r.x, r.y), fmaxf(r.z, r.w)));
    }
    // wave32 reduction
    for (int off = 16; off > 0; off >>= 1) {
        mn = fminf(mn, __shfl_xor(mn, off, 32));
        mx = fmaxf(mx, __shfl_xor(mx, off, 32));
    }
    __shared__ float smn[8], smx[8];
    unsigned int wave = threadIdx.x >> 5;
    unsigned int lane = threadIdx.x & 31u;
    if (lane == 0) { smn[wave] = mn; smx[wave] = mx; }
    __syncthreads();
    if (threadIdx.x == 0) {
        for (int w = 1; w < 8; ++w) { mn = fminf(mn, smn[w]); mx = fmaxf(mx, smx[w]); }
        // all r positive -> float order == uint-bit order
        atomicMin(&ws[0], __float_as_uint(mn));
        atomicMax(&ws[1], __float_as_uint(mx));
    }
}

__global__ __launch_bounds__(256) void k_hist(const float* __restrict__ y,
                                              unsigned int* __restrict__ ws,
                                              unsigned int total4) {
    __shared__ unsigned int lh[8][NBINS];   // per-wave privatized histograms
    unsigned int wave = threadIdx.x >> 5;
    for (int j = threadIdx.x; j < 8 * NBINS; j += blockDim.x)
        ((unsigned int*)lh)[j] = 0u;
    __syncthreads();

    float mn = __uint_as_float(ws[0]);
    float mx = __uint_as_float(ws[1]);
    float scale = (mx > mn) ? ((float)NBINS / (mx - mn)) : 0.0f;

    unsigned int tid = blockIdx.x * blockDim.x + threadIdx.x;
    unsigned int stride = gridDim.x * blockDim.x;
    for (unsigned int i = tid; i < total4; i += stride) {
        float4 r = load_r4(y, i, stride);
        int b0 = min(max((int)((r.x - mn) * scale), 0), NBINS - 1);
        int b1 = min(max((int)((r.y - mn) * scale), 0), NBINS - 1);
        int b2 = min(max((int)((r.z - mn) * scale), 0), NBINS - 1);
        int b3 = min(max((int)((r.w - mn) * scale), 0), NBINS - 1);
        atomicAdd(&lh[wave][b0], 1u);
        atomicAdd(&lh[wave][b1], 1u);
        atomicAdd(&lh[wave][b2], 1u);
        atomicAdd(&lh[wave][b3], 1u);
    }
    __syncthreads();
    for (int b = threadIdx.x; b < NBINS; b += blockDim.x) {
        unsigned int s = 0;
        for (int w = 0; w < 8; ++w) s += lh[w][b];
        if (s) atomicAdd(&ws[2 + b], s);
    }
}

// One wave (32 lanes, EXEC all ones -> WMMA legal). Each lane owns 2 of 64
// padded bins per the 4x16 B-matrix layout used by wave_sum64_wmma.
__global__ void k_final(const unsigned int* __restrict__ ws,
                        const float* __restrict__ sigma,
                        float* __restrict__ out,
                        float ncount) {
    int t = threadIdx.x;                       // 0..31
    float mn = __uint_as_float(ws[0]);
    float mx = __uint_as_float(ws[1]);
    float width = (mx - mn) * (1.0f / (float)NBINS);
    if (!(width > 0.0f)) width = 1.0f;

    int i0 = (t < 16) ? t : (t + 16);          // slots {t, t+16} / {t+16, t+32}
    int i1 = i0 + 16;

    float cnt0 = (i0 < NBINS) ? (float)ws[2 + i0] : 0.0f;
    float cnt1 = (i1 < NBINS) ? (float)ws[2 + i1] : 0.0f;

    // density histogram: counts / (N * bin_width)
    float inv = 1.0f / (ncount * width);
    float hv0 = cnt0 * inv;
    float hv1 = cnt1 * inv;

    float total = wave_sum64_wmma(hv0, hv1);   // WMMA reduction #1
    float denom = 1.0f / (total + KEPS);
    float hn0 = hv0 * denom;
    float hn1 = hv1 * denom;

    float s = fminf(fmaxf(sigma[0], 0.1f), 100.0f);
    float inv_s2 = 1.0f / (s * s);
    float c0 = mn + ((float)i0 + 0.5f) * width;
    float c1 = mn + ((float)i1 + 0.5f) * width;
    float pdf0 = c0 * inv_s2 * expf(-0.5f * c0 * c0 * inv_s2) + KEPS;
    float pdf1 = c1 * inv_s2 * expf(-0.5f * c1 * c1 * inv_s2) + KEPS;

    float t0 = (i0 < NBINS) ? hn0 * logf((hn0 + KEPS) / pdf0) : 0.0f;
    float t1 = (i1 < NBINS) ? hn1 * logf((hn1 + KEPS) / pdf1) : 0.0f;

    float kl = wave_sum64_wmma(t0, t1);        // WMMA reduction #2
    if (t == 0) out[0] = kl;
}

extern "C" void kernel_launch(void* const* d_in, const int* in_sizes, int n_in,
                              void* d_out, int out_size, void* d_ws, size_t ws_size,
                              hipStream_t stream) {
    const float* y = (const float*)d_in[0];      // [64,2,512,512] f32
    const float* sigma = (const float*)d_in[1];  // [1] f32
    float* out = (float*)d_out;                  // scalar f32
    unsigned int* ws = (unsigned int*)d_ws;

    unsigned int total = (unsigned int)in_sizes[0];   // 33,554,432
    unsigned int pixels = total / 2u;                 // 16,777,216 r-values
    unsigned int total4 = pixels / 4u;                // float4 groups

    k_init<<<1, 64, 0, stream>>>(ws);
    k_minmax<<<2048, 256, 0, stream>>>(y, ws, total4);
    k_hist<<<2048, 256, 0, stream>>>(y, ws, total4);
    k_final<<<1, 32, 0, stream>>>(ws, sigma, out, (float)pixels);
}